// MultiStageFIRFilter_29824252903953
// MI455X (gfx1250) — compile-verified
//
#include <hip/hip_runtime.h>
#include <stdint.h>

// ---------------- problem constants (match reference) ----------------
#define B_DIM    4
#define T_DIM    16384
#define K_TAPS   25            // M_IN + 1
#define ORD      20            // TAYLOR_ORDER
// ---------------- tiling ----------------
#define TILE        512
#define HALO        (24 * ORD)        // 480: max lookback after 20 stages
#define TOT         (TILE + HALO)     // 992 samples staged per workgroup
#define NTHREADS    256               // 8 wave32 waves
#define TILES_PER_B (T_DIM / TILE)    // 32
#define MC_FLOATS   (TOT * K_TAPS)    // 24800
#define SMEM_FLOATS (MC_FLOATS + 2 * TOT)
#define SMEM_BYTES  (SMEM_FLOATS * 4) // 107136 B < 320KB CDNA5 LDS

#ifndef __has_builtin
#define __has_builtin(x) 0
#endif

#if defined(__HIP_DEVICE_COMPILE__) && \
    __has_builtin(__builtin_amdgcn_tensor_load_to_lds) && \
    __has_builtin(__builtin_amdgcn_s_wait_tensorcnt)
#define USE_TDM 1
#else
#define USE_TDM 0
#endif

#if USE_TDM
typedef unsigned int u32x4 __attribute__((ext_vector_type(4)));
typedef int          i32x4 __attribute__((ext_vector_type(4)));
typedef int          i32x8 __attribute__((ext_vector_type(8)));

// Flat 1-D TDM copy of nelem f32 elements, global -> LDS.
// D# group0: count=1 | lds_addr | global_addr[56:0] | type=2 @ [127:126]
// D# group1: data_size=2(4B); tensor_dim0=nelem; tensor_dim1=1;
//            tile_dim0=nelem (<=65535); tile_dim1/2=0 (1-D); stride0=nelem.
__device__ __forceinline__ void tdm_load_1d(void* lds_dst, const void* gsrc,
                                            unsigned nelem_f32) {
  unsigned long long ga = (unsigned long long)(uintptr_t)gsrc;
  unsigned lds = (unsigned)(uintptr_t)lds_dst;   // low 32 bits of flat LDS addr
  u32x4 g0;
  g0[0] = 1u;                                         // count=1, user mode
  g0[1] = lds;                                        // lds_addr (bytes)
  g0[2] = (unsigned)(ga & 0xFFFFFFFFull);             // global_addr[31:0]
  g0[3] = (unsigned)((ga >> 32) & 0x01FFFFFFull)      // global_addr[56:32]
        | 0x80000000u;                                // type=2 ("image")
  i32x8 g1;
  g1[0] = (int)0x00020000u;                           // data_size=2 (4B), mask=0
  g1[1] = (int)((nelem_f32 & 0xFFFFu) << 16);         // tensor_dim0[15:0] @63:48
  g1[2] = (int)(((nelem_f32 >> 16) & 0xFFFFu)         // tensor_dim0[31:16]
        | (1u << 16));                                // tensor_dim1 = 1
  g1[3] = (int)((nelem_f32 & 0xFFFFu) << 16);         // tile_dim0 @127:112
  g1[4] = 0;                                          // tile_dim1=0, tile_dim2=0
  g1[5] = (int)nelem_f32;                             // tensor_dim0_stride lo32
  g1[6] = 0;
  g1[7] = 0;
  i32x4 gz4 = (i32x4){0, 0, 0, 0};                    // groups 2/3 unused (1-D)
  i32x8 gz8 = (i32x8){0, 0, 0, 0, 0, 0, 0, 0};        // extra group (6-arg form)
  // amdgpu-toolchain (clang-23) 6-arg form: (g0, g1, g2, g3, g4, cpol)
  __builtin_amdgcn_tensor_load_to_lds(g0, g1, gz4, gz4, gz8, 0);
}
#endif

__global__ void __launch_bounds__(NTHREADS, 1)
MultiStageFIRFilter_29824252903953_kernel(const float* __restrict__ x,
                                          const float* __restrict__ mc,
                                          float* __restrict__ y) {
  extern __shared__ float smem[];
  float* mcl  = smem;               // TOT * 25 taps (row stride 25: odd ->
                                    // conflict-free across stride-2 lanes)
  float* bufA = smem + MC_FLOATS;   // ping
  float* bufB = bufA + TOT;         // pong

  const int  tid   = threadIdx.x;
  const int  b     = blockIdx.x / TILES_PER_B;
  const int  tile  = blockIdx.x % TILES_PER_B;
  const int  t0    = tile * TILE;
  const bool first = (tile == 0);
  const int  lo_t  = first ? HALO : 0;            // first tile has no real halo
  const long long gt0 = (long long)b * T_DIM + t0 - HALO; // global t of local 0

  // ---------------- stage x tile (+halo) and mc tile into LDS ----------------
#if USE_TDM
  if (tid < 32) {  // one wave32 issues both TDM descriptors (EXEC all-ones)
    tdm_load_1d(&mcl[lo_t * K_TAPS],
                mc + (gt0 * K_TAPS + (long long)lo_t * K_TAPS),
                (unsigned)((TOT - lo_t) * K_TAPS));
    tdm_load_1d(&bufA[lo_t], x + (gt0 + lo_t), (unsigned)(TOT - lo_t));
    __builtin_amdgcn_s_wait_tensorcnt(0);
  }
#else
  for (int i = lo_t + tid; i < TOT; i += NTHREADS) bufA[i] = x[gt0 + i];
  for (int j = lo_t * K_TAPS + tid; j < MC_FLOATS; j += NTHREADS)
    mcl[j] = mc[gt0 * K_TAPS + j];
#endif
  // first tile: samples at global t<0 are zero; only [HALO-24, HALO) is read
  if (first && tid < 24) {
    bufA[HALO - 24 + tid] = 0.0f;
    bufB[HALO - 24 + tid] = 0.0f;
  }
  __syncthreads();

  // each thread owns 2 consecutive output samples: local [base, base+1]
  const int base = HALO + 2 * tid;
  float acc0 = bufA[base];          // y starts as x
  float acc1 = bufA[base + 1];

  float* xa = bufA;
  float* xb = bufB;
  for (int a = 1; a <= ORD; ++a) {
    const float inva = 1.0f / (float)a;

    // halo region [24*a, HALO): valid zone shrinks 24/stage; skipped for tile 0
    if (!first) {
      for (int i = 24 * a + tid; i < HALO; i += NTHREADS) {
        const float* crow = &mcl[i * K_TAPS];
        float s = 0.0f;
#pragma unroll
        for (int k = 0; k < K_TAPS; ++k) s = fmaf(crow[k], xa[i - k], s);
        xb[i] = s * inva;
      }
    }

    // output region: register-cached 26-float window covers both samples
    float w[26];
#pragma unroll
    for (int j = 0; j < 26; ++j) w[j] = xa[base - 24 + j];

    const float* c0 = &mcl[base * K_TAPS];
    const float* c1 = c0 + K_TAPS;
    float s0 = 0.0f, s1 = 0.0f;
#pragma unroll
    for (int k = 0; k < K_TAPS; ++k) s0 = fmaf(c0[k], w[24 - k], s0);
#pragma unroll
    for (int k = 0; k < K_TAPS; ++k) s1 = fmaf(c1[k], w[25 - k], s1);
    s0 *= inva;
    s1 *= inva;
    xb[base]     = s0;
    xb[base + 1] = s1;
    acc0 += s0;
    acc1 += s1;

    __syncthreads();          // one barrier per Taylor stage
    float* t = xa; xa = xb; xb = t;
  }

  // coalesced 8B store of the two owned samples
  float2 o; o.x = acc0; o.y = acc1;
  *reinterpret_cast<float2*>(y + (size_t)b * T_DIM + t0 + 2 * tid) = o;
}

extern "C" void kernel_launch(void* const* d_in, const int* in_sizes, int n_in,
                              void* d_out, int out_size, void* d_ws,
                              size_t ws_size, hipStream_t stream) {
  (void)in_sizes; (void)n_in; (void)out_size; (void)d_ws; (void)ws_size;
  const float* x  = (const float*)d_in[0];
  const float* mc = (const float*)d_in[1];
  float* y = (float*)d_out;

  // 107 KB dynamic LDS (> default 64 KB cap; CDNA5 WGP has 320 KB)
  (void)hipFuncSetAttribute(
      reinterpret_cast<const void*>(MultiStageFIRFilter_29824252903953_kernel),
      hipFuncAttributeMaxDynamicSharedMemorySize, SMEM_BYTES);

  dim3 grid(B_DIM * TILES_PER_B);   // 128 workgroups, fully independent
  dim3 block(NTHREADS);             // 8 wave32 waves
  hipLaunchKernelGGL(MultiStageFIRFilter_29824252903953_kernel, grid, block,
                     SMEM_BYTES, stream, x, mc, y);
}